// GeometricInteractionLayer3D_13950053777574
// MI455X (gfx1250) — compile-verified
//
#include <hip/hip_runtime.h>

#define DIM       2048
#define NC        12
#define NC2       144           // NC*NC
#define TOKS      16            // tokens per block (one WMMA M-tile)
#define XROW      2052          // padded LDS row (floats): 2052 % 64 == 4 -> conflict free
#define IROW      148           // padded inter row (floats): 148 % 64 == 20 -> conflict free
#define NTHREADS  512           // 16 waves of 32
#define NWAVES    16
#define KPW       (DIM / NWAVES)   // 128 K-elements per wave in projection phase

typedef __attribute__((ext_vector_type(2))) float v2f;
typedef __attribute__((ext_vector_type(8))) float v8f;

__global__ __launch_bounds__(NTHREADS)
void geo_interact_ln_kernel(const float* __restrict__ x,
                            const float* __restrict__ Wp, const float* __restrict__ bp,
                            const float* __restrict__ Wv, const float* __restrict__ bv,
                            const float* __restrict__ Wi, const float* __restrict__ bi,
                            const float* __restrict__ gamma, const float* __restrict__ beta,
                            float* __restrict__ out)
{
    extern __shared__ float smem[];
    float* sX      = smem;                        // 16 * 2052 floats
    float* sRegion = sX + TOKS * XROW;            // union: sRed (6144) / sInter (16*148)
    float* sRed    = sRegion;                     // 16 waves * 2 mats * 16 tok * 12 comp
    float* sInter  = sRegion;                     // live only after reduction barrier
    float* sPos    = sRegion + NWAVES * 2 * TOKS * NC;  // 16*12
    float* sVel    = sPos + TOKS * NC;                  // 16*12

    const int tid    = threadIdx.x;
    const int token0 = blockIdx.x * TOKS;
    const int wave   = tid >> 5;
    const int lane   = tid & 31;
    const int l16    = lane & 15;
    const int lhalf  = lane >> 4;          // 0: K=k0,k0+1   1: K=k0+2,k0+3

    // ---- Phase 1: stage x tile [16 x 2048] into LDS (float4, padded rows) ----
    {
        const float4* xg = (const float4*)(x + (size_t)token0 * DIM);
        for (int f = tid; f < TOKS * (DIM / 4); f += NTHREADS) {
            int row = f >> 9;            // 512 float4 per row
            int c4  = f & 511;
            float4 v = xg[(size_t)row * (DIM / 4) + c4];
            *(float4*)(sX + row * XROW + c4 * 4) = v;
        }
    }
    __syncthreads();

    // ---- Phase 2: pos/vel projections via WMMA, K split across 16 waves ----
    // C[M=token][N=comp] += x[M][k] * W[k][N]; each wave covers 128 K-values.
    // B columns 12..15 are garbage duplicates of column 11 (clamped address:
    // branch-free, always in bounds). Each C column depends only on its own B
    // column, and we never store C columns >= 12, so no zeroing is needed.
    {
        const int kbase = wave * KPW;
        const int cls   = (l16 < NC) ? l16 : (NC - 1);   // clamped comp index
        v8f accP = {}, accV = {};
        #pragma unroll
        for (int k0 = 0; k0 < KPW; k0 += 4) {
            const int kk = kbase + k0 + (lhalf << 1);
            v2f a = *(const v2f*)(sX + l16 * XROW + kk);   // A[M=l16][kk..kk+1]
            v2f bpv, bvv;
            bpv.x = Wp[kk * NC + cls];                     // B[kk][comp]
            bpv.y = Wp[(kk + 1) * NC + cls];
            bvv.x = Wv[kk * NC + cls];
            bvv.y = Wv[(kk + 1) * NC + cls];
            accP = __builtin_amdgcn_wmma_f32_16x16x4_f32(
                       false, a, false, bpv, (short)0, accP, false, false);
            accV = __builtin_amdgcn_wmma_f32_16x16x4_f32(
                       false, a, false, bvv, (short)0, accV, false, false);
        }
        if (l16 < NC) {
            #pragma unroll
            for (int r = 0; r < 8; ++r) {
                int t = r + lhalf * 8;                     // C/D: vgpr r -> M = r + 8*lhalf
                sRed[((wave * 2 + 0) * TOKS + t) * NC + l16] = accP[r];
                sRed[((wave * 2 + 1) * TOKS + t) * NC + l16] = accV[r];
            }
        }
    }
    __syncthreads();

    // ---- Phase 2b: reduce 16 wave-partials, add biases -> sPos / sVel ----
    if (tid < 2 * TOKS * NC) {                 // 384 tasks: (mat, token, comp)
        int s  = tid / (TOKS * NC);
        int r2 = tid - s * (TOKS * NC);
        int t  = r2 / NC;
        int c  = r2 - t * NC;
        float acc = (s == 0) ? bp[c] : bv[c];
        #pragma unroll
        for (int w = 0; w < NWAVES; ++w)
            acc += sRed[((w * 2 + s) * TOKS + t) * NC + c];
        if (s == 0) sPos[t * NC + c] = acc; else sVel[t * NC + c] = acc;
    }
    __syncthreads();

    // ---- Phase 3: outer product -> inter tile [16 x 144] (overwrites sRed) ----
    for (int e = tid; e < TOKS * NC2; e += NTHREADS) {
        int t = e / NC2;
        int k = e - t * NC2;
        int i = k / NC;
        int j = k - i * NC;
        sInter[t * IROW + k] = sPos[t * NC + i] * sVel[t * NC + j];
    }
    __syncthreads();

    // ---- Phase 4: WMMA GEMM  inter[16x144] @ Wi[144x2048]; y = x + inter + bi -> sX ----
    {
        const float* Arow = sInter + l16 * IROW;
        for (int tile = 0; tile < 8; ++tile) {           // 16 waves * 8 tiles = 128 N-tiles
            const int n0  = (wave * 8 + tile) * 16;
            const int nn  = n0 + l16;
            const float biv = bi[nn];
            const float* Bcol = Wi + nn;                 // Wi is [144, 2048] row-major
            v8f acc = {};
            #pragma unroll
            for (int k0 = 0; k0 < NC2; k0 += 4) {
                const int kk = k0 + (lhalf << 1);
                v2f a = *(const v2f*)(Arow + kk);        // A[M=l16][kk..kk+1]
                v2f b;
                b.x = Bcol[(size_t)kk * DIM];            // B[kk][nn]
                b.y = Bcol[(size_t)(kk + 1) * DIM];      // B[kk+1][nn]
                acc = __builtin_amdgcn_wmma_f32_16x16x4_f32(
                          false, a, false, b, (short)0, acc, false, false);
            }
            const int mbase = lhalf * 8;
            #pragma unroll
            for (int r = 0; r < 8; ++r) {
                sX[(mbase + r) * XROW + nn] += acc[r] + biv;
            }
        }
    }
    __syncthreads();

    // ---- Phase 5: LayerNorm — wave w handles token w ----
    {
        const float* yr = sX + wave * XROW;
        float s = 0.f, s2 = 0.f;
        for (int i = lane; i < DIM; i += 32) {
            float v = yr[i];
            s += v; s2 += v * v;
        }
        #pragma unroll
        for (int off = 16; off > 0; off >>= 1) {
            s  += __shfl_xor(s,  off, 32);
            s2 += __shfl_xor(s2, off, 32);
        }
        const float mean = s  * (1.0f / DIM);
        const float var  = s2 * (1.0f / DIM) - mean * mean;
        const float rstd = rsqrtf(var + 1e-5f);
        float* og = out + (size_t)(token0 + wave) * DIM;
        for (int i = lane; i < DIM; i += 32) {
            og[i] = (yr[i] - mean) * rstd * gamma[i] + beta[i];
        }
    }
}

extern "C" void kernel_launch(void* const* d_in, const int* in_sizes, int n_in,
                              void* d_out, int out_size, void* d_ws, size_t ws_size,
                              hipStream_t stream) {
    const float* x     = (const float*)d_in[0];
    const float* Wp    = (const float*)d_in[1];
    const float* bp    = (const float*)d_in[2];
    const float* Wv    = (const float*)d_in[3];
    const float* bv    = (const float*)d_in[4];
    const float* Wi    = (const float*)d_in[5];
    const float* bi    = (const float*)d_in[6];
    const float* gamma = (const float*)d_in[7];
    const float* beta  = (const float*)d_in[8];
    float* out = (float*)d_out;

    const int tokens = in_sizes[0] / DIM;           // 8 * 4096 = 32768
    const int blocks = tokens / TOKS;               // 2048

    constexpr int SMEM_FLOATS = TOKS * XROW                 // x tile       32832
                              + NWAVES * 2 * TOKS * NC      // sRed union    6144
                              + 2 * TOKS * NC;              // sPos/sVel      384
    constexpr int SMEM_BYTES  = SMEM_FLOATS * (int)sizeof(float);  // 157440

    hipFuncSetAttribute(reinterpret_cast<const void*>(geo_interact_ln_kernel),
                        hipFuncAttributeMaxDynamicSharedMemorySize, SMEM_BYTES);

    geo_interact_ln_kernel<<<blocks, NTHREADS, SMEM_BYTES, stream>>>(
        x, Wp, bp, Wv, bv, Wi, bi, gamma, beta, out);
}